// LayerSample_25744033972457
// MI455X (gfx1250) — compile-verified
//
#include <hip/hip_runtime.h>
#include <hip/hip_bf16.h>

#define BLK   256
#define TILE  4096
#define NIT   (TILE / BLK)
#define NWAVE (BLK / 32)

// ---------------- gfx1250 feature helpers (guarded, with fallbacks) --------

__device__ __forceinline__ unsigned ballot32(bool p) {
#if __has_builtin(__builtin_amdgcn_ballot_w32)
  return __builtin_amdgcn_ballot_w32(p);          // wave32-native ballot
#else
  return (unsigned)__ballot((int)p);
#endif
}

// Builtin signature (from clang diagnostic): params are typed v2i pointers in
// global/LDS address spaces.
typedef int v2i_t __attribute__((vector_size(8)));
typedef __attribute__((address_space(1))) v2i_t* gptr_v2i;
typedef __attribute__((address_space(3))) v2i_t* lptr_v2i;

#if __has_builtin(__builtin_amdgcn_global_load_async_to_lds_b64)
#define HAVE_ASYNC_LDS 1
__device__ __forceinline__ void async_copy_b64(const void* g, void* l) {
  __builtin_amdgcn_global_load_async_to_lds_b64(
      (gptr_v2i)g, (lptr_v2i)l, 0, 0);
}
#else
#define HAVE_ASYNC_LDS 0
__device__ __forceinline__ void async_copy_b64(const void* g, void* l) {
  *(volatile long long*)l = *(const long long*)g; // synchronous fallback
}
#endif

template <int N>
__device__ __forceinline__ void wait_async() {
#if HAVE_ASYNC_LDS
#if __has_builtin(__builtin_amdgcn_s_wait_asynccnt)
  __builtin_amdgcn_s_wait_asynccnt(N);
#else
  asm volatile("s_wait_asynccnt %0" ::"i"(N) : "memory");
#endif
#endif
}

// ---------------- kernel: stable compacting scatter ------------------------
// (defined first so the disasm snippet shows the async-to-LDS pipeline)
// edge_index rows staged global->LDS with gfx1250 async copies, double
// buffered; wave32 ballot rank + cross-wave LDS scan gives stable order.

__global__ void __launch_bounds__(BLK)
scatter_kernel(const long long* __restrict__ ei,
               const float* __restrict__ ea,
               const float* __restrict__ srate,
               const unsigned char* __restrict__ sampled,
               const int* __restrict__ blkoff,
               long long E, float* __restrict__ out) {
  __shared__ long long ssrc[2][BLK];
  __shared__ long long sdst[2][BLK];
  __shared__ int wcnt[NWAVE];

  const int  lane = threadIdx.x & 31;
  const int  wid  = threadIdx.x >> 5;
  const long long base = (long long)blockIdx.x * TILE;
  const float invrate  = 1.0f + __expf(-srate[0]);   // 1/sigmoid(s)

  float* out_src = out;
  float* out_dst = out + E;
  float* out_att = out + 2 * E;

  auto issue = [&](int it, int buf) {
    long long e  = base + (long long)it * BLK + threadIdx.x;
    long long ec = (e < E) ? e : (E - 1);            // clamp: fixed async count
    async_copy_b64(ei + ec,     &ssrc[buf][threadIdx.x]);
    async_copy_b64(ei + E + ec, &sdst[buf][threadIdx.x]);
  };

  issue(0, 0);
  long long running = blkoff[blockIdx.x];

  for (int it = 0; it < NIT; ++it) {
    if (it + 1 < NIT) {
      issue(it + 1, (it + 1) & 1);   // pipeline next chunk
      wait_async<2>();               // current chunk's 2 loads complete
    } else {
      wait_async<0>();
    }

    long long e = base + (long long)it * BLK + threadIdx.x;
    bool keep = false;
    long long s = 0, d = 0;
    if (e < E) {
      s = ssrc[it & 1][threadIdx.x];
      d = sdst[it & 1][threadIdx.x];
      keep = (sampled[s] != 0) & (sampled[d] != 0);
    }

    unsigned bal = ballot32(keep);
    if (lane == 0) wcnt[wid] = __popc(bal);
    __syncthreads();

    int prefix = 0, tot = 0;
    for (int w = 0; w < NWAVE; ++w) {
      int c = wcnt[w];
      if (w < wid) prefix += c;
      tot += c;
    }

    if (keep) {
      long long p = running + prefix + __popc(bal & ((1u << lane) - 1u));
      out_src[p] = (float)s;
      out_dst[p] = (float)d;
      __builtin_prefetch(ea + e + BLK, 0, 3);        // WGP-scope prefetch
      out_att[p] = ea[e] * invrate;
    }
    running += tot;
    __syncthreads();                                 // wcnt reuse guard
  }
}

// ---------------- kernel: per-block keep counts ----------------------------

__global__ void __launch_bounds__(BLK)
count_kernel(const long long* __restrict__ ei,
             const unsigned char* __restrict__ sampled,
             long long E, int* __restrict__ blkcnt) {
  __shared__ int wsum[NWAVE];
  const int  lane = threadIdx.x & 31;
  const int  wid  = threadIdx.x >> 5;
  const long long base = (long long)blockIdx.x * TILE;

  int mycnt = 0;
  for (int it = 0; it < NIT; ++it) {
    long long e = base + (long long)it * BLK + threadIdx.x;
    // prefetch next iteration of both int64 rows (speculative, OOB-safe)
    __builtin_prefetch(ei + e + BLK, 0, 3);
    __builtin_prefetch(ei + E + e + BLK, 0, 3);
    bool keep = false;
    if (e < E) {
      long long s = ei[e];
      long long d = ei[E + e];
      keep = (sampled[s] != 0) & (sampled[d] != 0);
    }
    unsigned bal = ballot32(keep);
    if (lane == 0) mycnt += __popc(bal);
  }
  if (lane == 0) wsum[wid] = mycnt;
  __syncthreads();
  if (threadIdx.x == 0) {
    int t = 0;
    for (int w = 0; w < NWAVE; ++w) t += wsum[w];
    blkcnt[blockIdx.x] = t;
  }
}

// ---------------- kernel: exclusive scan of block counts -------------------

__global__ void __launch_bounds__(BLK)
scan_kernel(const int* __restrict__ cnt, int* __restrict__ off,
            int* __restrict__ total, float* __restrict__ out_count, int nb) {
  __shared__ int tmp[BLK];
  const int per   = (nb + BLK - 1) / BLK;
  const int begin = threadIdx.x * per;

  int local = 0;
  for (int i = 0; i < per; ++i) {
    int j = begin + i;
    if (j < nb) local += cnt[j];
  }
  tmp[threadIdx.x] = local;
  __syncthreads();
  // Hillis-Steele inclusive scan over 256 thread partials
  for (int d = 1; d < BLK; d <<= 1) {
    int t = (threadIdx.x >= d) ? tmp[threadIdx.x - d] : 0;
    __syncthreads();
    tmp[threadIdx.x] += t;
    __syncthreads();
  }
  int acc = tmp[threadIdx.x] - local;  // exclusive prefix for this thread
  for (int i = 0; i < per; ++i) {
    int j = begin + i;
    if (j < nb) { off[j] = acc; acc += cnt[j]; }
  }
  if (threadIdx.x == BLK - 1) {
    *total     = tmp[BLK - 1];
    *out_count = (float)tmp[BLK - 1];
  }
}

// ---------------- kernel: fill padding region [count, E) -------------------

__global__ void __launch_bounds__(BLK)
pad_kernel(float* __restrict__ out, const int* __restrict__ total, long long E) {
  long long j = (long long)blockIdx.x * BLK + threadIdx.x;
  int cnt = *total;                                  // uniform scalar load
  if (j < E && j >= (long long)cnt) {
    out[j]         = -1.0f;
    out[E + j]     = -1.0f;
    out[2 * E + j] =  0.0f;
  }
}

// ---------------- host launcher --------------------------------------------

extern "C" void kernel_launch(void* const* d_in, const int* in_sizes, int n_in,
                              void* d_out, int out_size, void* d_ws, size_t ws_size,
                              hipStream_t stream) {
  // inputs: 0=x (unused), 1=edge_index int64[2E], 2=edge_attr f32[E],
  //         3=sample_rate f32[1], 4=sampled bool[N]
  const long long*     ei      = (const long long*)d_in[1];
  const float*         ea      = (const float*)d_in[2];
  const float*         srate   = (const float*)d_in[3];
  const unsigned char* sampled = (const unsigned char*)d_in[4];

  const long long E  = (long long)in_sizes[2];
  const int       nb = (int)((E + TILE - 1) / TILE);

  float* out    = (float*)d_out;
  int*   ws     = (int*)d_ws;
  int*   blkcnt = ws;
  int*   blkoff = ws + nb;
  int*   total  = ws + 2 * nb;

  count_kernel  <<<nb, BLK, 0, stream>>>(ei, sampled, E, blkcnt);
  scan_kernel   <<<1,  BLK, 0, stream>>>(blkcnt, blkoff, total, out + 3 * E, nb);
  scatter_kernel<<<nb, BLK, 0, stream>>>(ei, ea, srate, sampled, blkoff, E, out);
  pad_kernel    <<<(unsigned)((E + BLK - 1) / BLK), BLK, 0, stream>>>(out, total, E);
}